// Attention_Luong_no_batch_56083682951387
// MI455X (gfx1250) — compile-verified
//
#include <hip/hip_runtime.h>
#include <math.h>
#include <stdint.h>

typedef __attribute__((ext_vector_type(2))) float v2f;
typedef __attribute__((ext_vector_type(8))) float v8f;

#define H 1024
#define S 32768
#define KC 128                 // k-chunk per pipeline stage
#define ROWPITCH 132           // 132 mod 64 = 4 -> conflict-free A-fragment reads
#define WAVES 8
#define RPW 16                 // rows per wave
#define NBLK (S / (WAVES * RPW))   // 256 score blocks

// ---------------- Kernel A: v = W^T h  (W is [d,h] row-major) ----------------
__global__ __launch_bounds__(1024) void wvec_kernel(const float* __restrict__ W,
                                                    const float* __restrict__ hidden,
                                                    float* __restrict__ v) {
    __shared__ float part[16][64];
    const int tid   = threadIdx.x;
    const int col   = (blockIdx.x << 6) + (tid & 63);   // 64 columns per block
    const int slice = tid >> 6;                          // 16 i-slices of 64
    float s = 0.f;
    const int i0 = slice << 6;
#pragma unroll 8
    for (int i = i0; i < i0 + 64; ++i)
        s = fmaf(W[(size_t)i * H + col], hidden[i], s);
    part[slice][tid & 63] = s;
    __syncthreads();
    if (slice == 0) {
        float t = 0.f;
#pragma unroll
        for (int q = 0; q < 16; ++q) t += part[q][tid];  // fixed order: deterministic
        v[col] = t;
    }
}

// ---------------- Kernel B: scores = enc @ v -------------------------------
// Async global->LDS staging (ASYNCcnt double buffer) + V_WMMA_F32_16X16X4_F32.
__global__ __launch_bounds__(256) void scores_kernel(const float* __restrict__ enc,
                                                     const float* __restrict__ v,
                                                     float* __restrict__ scores,
                                                     float* __restrict__ blockMax) {
    __shared__ float vlds[H];                            // 4 KB
    __shared__ float albs[WAVES][2][RPW * ROWPITCH];     // ~135 KB, double-buffered
    __shared__ float sred[WAVES * RPW];                  // 128 block scores

    const int tid  = threadIdx.x;
    const int wave = tid >> 5;
    const int lane = tid & 31;

    for (int i = tid; i < H; i += 256) vlds[i] = v[i];
    __syncthreads();

    const int rowBase = blockIdx.x * (WAVES * RPW) + wave * RPW;
    const float* encBase = enc + (size_t)rowBase * H;

    // Wave-relative LDS byte offsets of the two staging buffers
    // (flat LDS aperture keeps the offset in addr[31:0]).
    const uint32_t ldsBufByte0 = (uint32_t)(uintptr_t)&albs[wave][0][0];
    const uint32_t ldsBufByte1 = (uint32_t)(uintptr_t)&albs[wave][1][0];

    v8f acc = {0.f, 0.f, 0.f, 0.f, 0.f, 0.f, 0.f, 0.f};

    const int lcol = lane << 2;            // lane covers 16B of each 512B row
    const int arow = lane & 15;            // A-fragment row (M)
    const int ksub = (lane >> 4) << 1;     // lanes 16-31 hold K=2,3

    // stage(kc, buf): 16 rows x 128 floats, straight into LDS, no VGPR staging
    auto stage = [&](int kcDst, uint32_t ldsBase) {
        const float* g0 = encBase + (size_t)kcDst * KC + lcol;
        const uint32_t l0 = ldsBase + (uint32_t)(lane << 4);
#pragma unroll
        for (int r = 0; r < RPW; ++r) {
            const float* ga = g0 + (size_t)r * H;
            const uint32_t la = l0 + (uint32_t)(r * (ROWPITCH * 4));
            asm volatile("global_load_async_to_lds_b128 %0, %1, off"
                         :: "v"(la), "v"(ga) : "memory");
        }
    };

    stage(0, ldsBufByte0);
    int buf = 0;

    for (int kc = 0; kc < H / KC; ++kc) {
        if (kc + 1 < H / KC) {
            stage(kc + 1, buf ? ldsBufByte0 : ldsBufByte1);
            // async loads retire in order: <=16 outstanding => current buf ready
            asm volatile("s_wait_asynccnt 0x10" ::: "memory");
        } else {
            asm volatile("s_wait_asynccnt 0x0" ::: "memory");
        }

        const float* a0 = buf ? (const float*)&albs[wave][1][arow * ROWPITCH + ksub]
                              : (const float*)&albs[wave][0][arow * ROWPITCH + ksub];
        const float* vk = &vlds[kc * KC + ksub];
#pragma unroll
        for (int k = 0; k < KC; k += 4) {
            v2f a = *(const v2f*)(a0 + k);   // A[M, k..k+3] fragment (2 VGPRs)
            v2f b = *(const v2f*)(vk + k);   // B[K, N]=v[k0+K] replicated over N
            acc = __builtin_amdgcn_wmma_f32_16x16x4_f32(
                false, a, false, b, (short)0, acc, false, false);
        }
        buf ^= 1;
    }

    // D: vgpr j = row j (lanes 0-15) / row j+8 (lanes 16-31); all N columns equal
    if ((lane & 15) == 0) {
        const int half = lane >> 4;
#pragma unroll
        for (int j = 0; j < 8; ++j) {
            const int r  = j + (half << 3);
            const float sc = acc[j];
            scores[rowBase + r] = sc;
            sred[wave * RPW + r] = sc;
        }
    }
    __syncthreads();
    for (int off = 64; off >= 1; off >>= 1) {
        if (tid < off) sred[tid] = fmaxf(sred[tid], sred[tid + off]);
        __syncthreads();
    }
    if (tid == 0) blockMax[blockIdx.x] = sred[0];
}

// ---------------- Kernel C: deterministic single-block softmax ---------------
__global__ __launch_bounds__(1024) void softmax_kernel(const float* __restrict__ scores,
                                                       const float* __restrict__ blockMax,
                                                       float* __restrict__ out) {
    __shared__ float red[1024];
    const int tid = threadIdx.x;

    red[tid] = (tid < NBLK) ? blockMax[tid] : -INFINITY;
    __syncthreads();
    for (int off = 512; off >= 1; off >>= 1) {
        if (tid < off) red[tid] = fmaxf(red[tid], red[tid + off]);
        __syncthreads();
    }
    const float M = red[0];
    __syncthreads();

    float sum = 0.f;
#pragma unroll
    for (int i = 0; i < S / 1024; ++i)
        sum += __expf(scores[tid + (i << 10)] - M);      // fixed order per thread
    red[tid] = sum;
    __syncthreads();
    for (int off = 512; off >= 1; off >>= 1) {
        if (tid < off) red[tid] += red[tid + off];        // fixed tree: deterministic
        __syncthreads();
    }
    const float inv = 1.0f / red[0];
#pragma unroll
    for (int i = 0; i < S / 1024; ++i)
        out[tid + (i << 10)] = __expf(scores[tid + (i << 10)] - M) * inv;
}

// ---------------- launcher ---------------------------------------------------
extern "C" void kernel_launch(void* const* d_in, const int* in_sizes, int n_in,
                              void* d_out, int out_size, void* d_ws, size_t ws_size,
                              hipStream_t stream) {
    (void)in_sizes; (void)n_in; (void)out_size; (void)ws_size;
    const float* hidden = (const float*)d_in[0];     // (1,1,H)
    const float* enc    = (const float*)d_in[1];     // (S,1,H)
    const float* W      = (const float*)d_in[2];     // (H,H)
    // d_in[3] = b : provably irrelevant (softmax shift invariance)

    float* ws       = (float*)d_ws;
    float* v        = ws;                 // H floats
    float* blockMax = ws + H;             // NBLK floats
    float* scores   = ws + H + NBLK;      // S floats
    float* out      = (float*)d_out;      // (1,1,S)

    wvec_kernel   <<<H / 64, 1024, 0, stream>>>(W, hidden, v);
    scores_kernel <<<NBLK,   256,  0, stream>>>(enc, v, scores, blockMax);
    softmax_kernel<<<1,      1024, 0, stream>>>(scores, blockMax, out);
}